// CustomStochastic_33500744908839
// MI455X (gfx1250) — compile-verified
//
#include <hip/hip_runtime.h>
#include <stdint.h>

// ---------------------------------------------------------------------------
// Stochastic LIF neuron batch: 65536 neurons x 1000 Euler-Maruyama steps.
// The 512 MB dW stream is staged through LDS with CDNA5 async-to-LDS b128
// copies, triple-buffered per wave (prefetch distance 2 chunks), tracked with
// ASYNCcnt. The rare spike-event path is hidden behind a wave-uniform
// __any() branch; a wave exits early once all 32 lanes have spiked.
//
// Roofline: 512 MB @ 23.3 TB/s ~= 22 us floor; common-path compute is now
// ~23 VALU + 1 trans per step/lane, keeping the kernel memory-limited.
// ---------------------------------------------------------------------------

#define NB          65536
#define NSTEPS      1000
#define CH          8                    // steps per chunk
#define NCHUNK      (NSTEPS / CH)        // 125
#define BLOCK       256
#define WAVES_PB    (BLOCK / 32)
#define STEP_BYTES  (NB * 8u)            // 524288 bytes between steps in dW
#define CHUNK_BYTES (CH * STEP_BYTES)
#define BUFB        (CH * 256u)          // 2048 B per buffer per wave
#define NBUF        3

#define DT        0.01f
#define ALPHA     0.03f
#define THRESH    1.0f
#define T1_SENT   10.0f
#define LOG2E     1.4426950408889634f

// One chunk = CH steps. Lanes 0-15 carry step 2j (16B = 2 neurons/lane),
// lanes 16-31 carry step 2j+1, so 4 x b128 move the whole 2 KB chunk.
__device__ __forceinline__ void issue_chunk(uint64_t gbase, unsigned goff,
                                            unsigned ldsoff) {
#pragma unroll
  for (int j = 0; j < CH / 2; ++j) {
    // GVS mode: global addr = SADDR(64) + VADDR(32). IOFFSET must stay 0:
    // the ISA adds it to BOTH the LDS and the global address.
    asm volatile("global_load_async_to_lds_b128 %0, %1, %2 offset:0"
                 :
                 : "v"(ldsoff + (unsigned)(j * 512u)),
                   "v"(goff + (unsigned)(2 * j) * STEP_BYTES),
                   "s"(gbase)
                 : "memory");
  }
}

__global__ __launch_bounds__(BLOCK) void sde_neuron_kernel(
    const float* __restrict__ Iin, const float* __restrict__ v0p,
    const float* __restrict__ i0p, const float* __restrict__ us0p,
    const float* __restrict__ dW, const float* __restrict__ mu,
    const float* __restrict__ sigma, float* __restrict__ out) {
  // Per-wave triple buffer: 3 bufs * CH steps * 32 lanes * 8 B = 6144 B.
  __shared__ __align__(16) float smem[WAVES_PB * NBUF * CH * 64];

  const int tid   = threadIdx.x;
  const int lane  = tid & 31;
  const int wave  = tid >> 5;
  const int gid   = (int)blockIdx.x * BLOCK + tid;
  const int wgid0 = gid & ~31;            // neuron id of lane 0 of this wave
  const int l16   = lane & 15;
  const int half  = lane >> 4;

  const float dt      = DT;
  const float sqrt_dt = sqrtf(DT);
  const float mu1 = mu[0], mu2 = mu[1];
  const float a1  = mu1 * dt;             // mu1*dt
  const float nb2 = -mu2 * dt;            // -mu2*dt
  const float c00 = sigma[0] * sqrt_dt, c01 = sigma[1] * sqrt_dt;
  const float c10 = sigma[2] * sqrt_dt, c11 = sigma[3] * sqrt_dt;
  const float Ic = Iin[gid];

  float v = v0p[gid];
  float i = i0p[gid];
  float u = us0p[gid];
  u = fminf(fmaxf(u, 1e-6f), 1.0f);
  float s = logf(u) - ALPHA;              // spike clock starts negative
  bool  spiked = false;
  float st = T1_SENT;
  float t  = 0.0f;

  // Low 32 bits of a __shared__ flat address == workgroup LDS byte address.
  const unsigned lds_wave = (unsigned)(uintptr_t)(&smem[wave * (NBUF * CH * 64)]);
  const unsigned lds_st   = lds_wave + (unsigned)(l16 * 16 + half * 256);
  const float*   rd_base  = &smem[wave * (NBUF * CH * 64) + lane * 2];

  const uint64_t gbase = (uint64_t)(uintptr_t)dW;
  // Per-lane global byte offset for (step half, neurons wgid0+2*l16..+1).
  unsigned goff = (unsigned)wgid0 * 8u + (unsigned)l16 * 16u +
                  (unsigned)half * STEP_BYTES;

  // Prime the pipeline: chunks 0 and 1 -> buffers 0 and 1.
  issue_chunk(gbase, goff, lds_st + 0u * BUFB);
  issue_chunk(gbase, goff + CHUNK_BYTES, lds_st + 1u * BUFB);
  goff += 2u * CHUNK_BYTES;

  unsigned proc_off  = 0;                 // byte offset of buffer to process
  unsigned issue_off = 2u * BUFB;         // byte offset of buffer to fill

#pragma unroll 1
  for (int c = 0; c < NCHUNK; ++c) {
    // Issue chunk c+2 and wait for chunk c. Async loads complete in order, so
    // "<= N youngest outstanding" pins the buffer we are about to read.
    if (c < NCHUNK - 2) {
      issue_chunk(gbase, goff, lds_st + issue_off);
      goff += CHUNK_BYTES;
      issue_off += BUFB;
      if (issue_off == NBUF * BUFB) issue_off = 0;
      asm volatile("s_wait_asynccnt 8" ::: "memory");
    } else if (c == NCHUNK - 2) {
      asm volatile("s_wait_asynccnt 4" ::: "memory");
    } else {
      asm volatile("s_wait_asynccnt 0" ::: "memory");
    }

    const float* rb = rd_base + (proc_off >> 2);
#pragma unroll
    for (int k = 0; k < CH; ++k) {
      const float2 nz = *(const float2*)(rb + k * 64);   // ds_load_b64

      // Diffusion with sqrt(dt) prefolded into sigma.
      const float nv = fmaf(c01, nz.y, c00 * nz.x);
      const float ni = fmaf(c11, nz.y, c10 * nz.x);
      // Drift + Euler-Maruyama (mu1*dt, -mu2*dt prefolded).
      const float ev = exp2f(v * LOG2E);                 // v_exp_f32
      const float ds = fmaxf(0.0f, ev * (v - THRESH));
      const float v_new = fmaf(a1, (i + Ic) - v, v) + nv;
      const float i_new = fmaf(nb2, i, i) + ni;          // i*(1-mu2*dt)+ni
      const float s_new = fmaf(ds, dt, s);

      // Event: s crosses 0 from below. Fires at most once per lane over the
      // whole run -> hide the interpolation behind a wave-uniform branch.
      const bool cross = (!spiked) && (s_new >= 0.0f);
      if (__builtin_expect(__any(cross), 0)) {
        const float denom = fmaxf(s_new - s, 1e-12f);
        const float frac  = (-s) * __builtin_amdgcn_rcpf(denom);  // v_rcp_f32
        if (cross) {
          st = fmaf(dt, frac, t);
          v  = fmaf(frac, v_new - v, v);
          i  = fmaf(frac, i_new - i, i);
          s  = 0.0f;
          spiked = true;
        } else if (!spiked) {
          v = v_new; i = i_new; s = s_new;
        }
      } else if (!spiked) {                // common path: 3 cndmasks
        v = v_new; i = i_new; s = s_new;
      }
      t += dt;
    }

    proc_off += BUFB;
    if (proc_off == NBUF * BUFB) proc_off = 0;

    // Frozen-after-spike: if every lane spiked, skip remaining loads+compute.
    if (__all(spiked)) break;
  }

  float4 r;
  r.x = v; r.y = i; r.z = s; r.w = st;
  *(float4*)(out + 4 * gid) = r;          // global_store_b128, coalesced
}

extern "C" void kernel_launch(void* const* d_in, const int* in_sizes, int n_in,
                              void* d_out, int out_size, void* d_ws,
                              size_t ws_size, hipStream_t stream) {
  const float* Iin   = (const float*)d_in[0];  // input_current [B]
  const float* v0    = (const float*)d_in[1];  // v0            [B]
  const float* i0    = (const float*)d_in[2];  // i0            [B]
  const float* us0   = (const float*)d_in[3];  // u_s0          [B]
  const float* dW    = (const float*)d_in[4];  // dW   [STEPS,B,2]
  const float* mu    = (const float*)d_in[5];  // mu            [2]
  const float* sigma = (const float*)d_in[6];  // sigma       [2,2]
  float* out = (float*)d_out;                  // [B,4]

  (void)in_sizes; (void)n_in; (void)out_size; (void)d_ws; (void)ws_size;

  hipLaunchKernelGGL(sde_neuron_kernel, dim3(NB / BLOCK), dim3(BLOCK), 0,
                     stream, Iin, v0, i0, us0, dW, mu, sigma, out);
}